// DepthCalibration_79783312490939
// MI455X (gfx1250) — compile-verified
//
#include <hip/hip_runtime.h>
#include <hip/hip_bf16.h>

typedef __attribute__((ext_vector_type(2))) float v2f;
typedef __attribute__((ext_vector_type(8))) float v8f;

#define NB 4096

// exp(-2*d2) = exp2(NEG2LOG2E * d2),  NEG2LOG2E = -2*log2(e)
#define NEG2LOG2E (-2.8853900817779268f)

__global__ __launch_bounds__(256) void DepthCalibration_kernel(
    const float* __restrict__ pred_depth,   // (8, 4096)
    const float* __restrict__ ray,          // (8, 4096, 3)
    const float* __restrict__ conv_w,       // (3,)
    const float* __restrict__ conv_b,       // (1,)
    const float* __restrict__ gscale,       // (1,)
    float* __restrict__ out)                // (8, 4096)
{
    const int lane  = threadIdx.x & 31;
    const int wave  = threadIdx.x >> 5;
    const int gw    = blockIdx.x * 8 + wave;     // 0..2047
    const int b     = gw >> 8;                   // batch 0..7
    const int rowbase = (gw & 255) * 16;         // 16-row block within batch

    const int j16  = lane & 15;
    const int half = lane >> 4;

    const float gs  = gscale[0];
    const float w0g = conv_w[0] * gs;            // fold global_scale into conv taps
    const float w1g = conv_w[1] * gs;
    const float w2g = conv_w[2] * gs;
    const float cb  = conv_b[0];

    const float* rayb = ray + (size_t)b * NB * 3;
    const float* pdb  = pred_depth + (size_t)b * NB;

    // ---- A fragment: 16x4 f32 rows of this row-block (K=3 dims + zero pad) ----
    // ISA layout: lane l -> M = l&15 ; K = {0,1} (lanes 0-15) / {2,3} (lanes 16-31)
    const int i = rowbase + j16;
    float a0 = rayb[i*3 + 0], a1 = rayb[i*3 + 1], a2 = rayb[i*3 + 2];
    float sq_i = a0*a0 + a1*a1 + a2*a2;
    v2f afrag;
    afrag.x = half ? a2   : a0;
    afrag.y = half ? 0.0f : a1;

    // sq (pre-scaled by -2*log2e) for the row each (VGPR r, half) maps to:
    // C/D layout row = rowbase + r + 8*half -> source lane r + 8*half
    float sqrowk[8];
    #pragma unroll
    for (int r = 0; r < 8; ++r)
        sqrowk[r] = __shfl(sq_i, r + half * 8, 32) * NEG2LOG2E;

    float acc[8];
    #pragma unroll
    for (int r = 0; r < 8; ++r) acc[r] = 0.0f;

    // ---- software-pipelined prefetch of tile 0 (branchless, clamped indices) ----
    int jn = j16;
    float nb0 = rayb[jn*3 + 0], nb1 = rayb[jn*3 + 1], nb2 = rayb[jn*3 + 2];
    float npm = pdb[max(jn - 1, 0)];
    float np0 = pdb[jn];
    float npp = pdb[min(jn + 1, NB - 1)];

    for (int t = 0; t < NB / 16; ++t) {
        const int j = t * 16 + j16;

        // consume prefetched tile data
        float r0 = nb0, r1 = nb1, r2 = nb2;
        float qm = npm, q0 = np0, qp = npp;

        // prefetch next tile (wrap last -> tile 0 to stay in-bounds; result unused)
        int tn = (t + 1) & (NB / 16 - 1);
        jn = tn * 16 + j16;
        nb0 = rayb[jn*3 + 0]; nb1 = rayb[jn*3 + 1]; nb2 = rayb[jn*3 + 2];
        npm = pdb[max(jn - 1, 0)];
        np0 = pdb[jn];
        npp = pdb[min(jn + 1, NB - 1)];

        // ---- B fragment: 4x16 f32, columns of this tile ----
        // layout mirrors A: VGPR0 = K0 (lanes 0-15) / K2 (lanes 16-31); VGPR1 = K1 / K3
        float sqcolk = (r0*r0 + r1*r1 + r2*r2) * NEG2LOG2E;
        v2f bfrag;
        bfrag.x = half ? r2   : r0;
        bfrag.y = half ? 0.0f : r1;

        // smoothed[j]: branchless zero-padding — zero the tap weight, not the load
        float c0 = (j > 0)      ? w0g : 0.0f;
        float c2 = (j < NB - 1) ? w2g : 0.0f;
        float sm = fmaf(c0, qm, fmaf(w1g, q0, fmaf(c2, qp, cb)));

        // inner products for the 16x16 tile: one WMMA, fp32, K=4
        v8f c = {};
        c = __builtin_amdgcn_wmma_f32_16x16x4_f32(
                /*neg_a=*/false, afrag, /*neg_b=*/false, bfrag,
                /*c_mod=*/(short)0, c, /*reuse_a=*/false, /*reuse_b=*/false);

        // arg = NEG2LOG2E*(sq_i + sq_j) + 2*log2e*2*inner ; d2>=0 clamp <=> min(arg,0)
        #pragma unroll
        for (int r = 0; r < 8; ++r) {
            float targ = fmaf(-2.0f * NEG2LOG2E, c[r], sqrowk[r] + sqcolk);
            targ = fminf(targ, 0.0f);
            float w = __builtin_amdgcn_exp2f(targ);   // v_exp_f32
            acc[r] = fmaf(w, sm, acc[r]);
        }
    }

    // Reduce over the 16 column-lanes in each half-wave (xor masks stay in-half)
    #pragma unroll
    for (int r = 0; r < 8; ++r) {
        float v = acc[r];
        v += __shfl_xor(v, 1, 32);
        v += __shfl_xor(v, 2, 32);
        v += __shfl_xor(v, 4, 32);
        v += __shfl_xor(v, 8, 32);
        acc[r] = v;
    }

    if (j16 == 0) {
        #pragma unroll
        for (int r = 0; r < 8; ++r) {
            int row = rowbase + r + half * 8;
            float v = fminf(fmaxf(acc[r], 0.1f), 100.0f);
            out[(size_t)b * NB + row] = v;
        }
    }
}

extern "C" void kernel_launch(void* const* d_in, const int* in_sizes, int n_in,
                              void* d_out, int out_size, void* d_ws, size_t ws_size,
                              hipStream_t stream) {
    const float* pred_depth = (const float*)d_in[0];
    const float* ray_3d     = (const float*)d_in[1];
    const float* conv_w     = (const float*)d_in[2];
    const float* conv_b     = (const float*)d_in[3];
    const float* gscale     = (const float*)d_in[4];
    float* out = (float*)d_out;

    // 8 batches * 256 row-blocks = 2048 waves; 8 waves (256 threads) per block
    dim3 block(256);
    dim3 grid(256);
    DepthCalibration_kernel<<<grid, block, 0, stream>>>(
        pred_depth, ray_3d, conv_w, conv_b, gscale, out);
}